// NetDesc_53755810677330
// MI455X (gfx1250) — compile-verified
//
#include <hip/hip_runtime.h>

#define NNODES 100000
#define NEDGES 3200000
#define HID 64
#define GDIM 32
#define NGRAPH 16

typedef float v2f __attribute__((ext_vector_type(2)));
typedef float v8f __attribute__((ext_vector_type(8)));

// ---------------------------------------------------------------------------
// Generic WMMA f32 GEMM:  out[., col_off + .] = act( X1/rowdiv @ W1 (+ X2@W2) + bias )
// X row-major [M x K] (row stride K), W row-major [K x N], out row stride ldo.
// Block: (N/16) waves; wave w computes the 16x16 tile at cols [16w,16w+16).
// Grid.x: ceil(M/16).  K, N, DUAL, ROWDIV, RELU are compile-time so the K-loop
// fully unrolls branch-free and LDS addressing folds into immediate offsets.
// ---------------------------------------------------------------------------
template <int K, int N, bool DUAL, bool ROWDIV, bool RELU>
__global__ __launch_bounds__((N / 16) * 32)
void wmma_gemm_kernel(const float* __restrict__ X1,
                      const float* __restrict__ rowdiv,
                      const float* __restrict__ W1,
                      const float* __restrict__ X2,
                      const float* __restrict__ W2,
                      const float* __restrict__ bias,   // may be null
                      float* __restrict__ out,
                      int M, int ldo, int col_off) {
  __shared__ float sA1[16 * K];
  __shared__ float sW1[K * N];
  __shared__ float sA2[DUAL ? 16 * K : 1];
  __shared__ float sW2[DUAL ? K * N : 1];

  const int tid   = threadIdx.x;
  const int lane  = tid & 31;
  const int wave  = tid >> 5;
  const int mbase = blockIdx.x * 16;

  // Cooperative loads into LDS (coalesced)
  for (int i = tid; i < 16 * K; i += (N / 16) * 32) {
    int r = i / K, c = i - r * K;
    int gr = mbase + r;
    float v1 = 0.0f, v2 = 0.0f;
    if (gr < M) {
      v1 = X1[(size_t)gr * K + c];
      if (ROWDIV) v1 /= fmaxf(rowdiv[gr], 1.0f);   // mean = agg / max(deg,1)
      if (DUAL) v2 = X2[(size_t)gr * K + c];
    }
    sA1[i] = v1;
    if (DUAL) sA2[i] = v2;
  }
  for (int i = tid; i < K * N; i += (N / 16) * 32) {
    sW1[i] = W1[i];
    if (DUAL) sW2[i] = W2[i];
  }
  __syncthreads();

  const int ml    = lane & 15;   // row (A) / col (B) within tile
  const int half  = lane >> 4;   // K-phase selector
  const int ncol0 = wave * 16;
  const int abase = ml * K + 2 * half;   // lane's A base
  const int bbase = ncol0 + ml;          // lane's B base (col)

  v8f c = {};
#pragma unroll
  for (int k0 = 0; k0 < K; k0 += 4) {
    const int ka = k0 + 2 * half;
    v2f a, b;
    a.x = sA1[abase + k0];
    a.y = sA1[abase + k0 + 1];
    b.x = sW1[ka * N + bbase];
    b.y = sW1[(ka + 1) * N + bbase];
    c = __builtin_amdgcn_wmma_f32_16x16x4_f32(false, a, false, b,
                                              (short)0, c, false, false);
    if (DUAL) {
      v2f a2, b2;
      a2.x = sA2[abase + k0];
      a2.y = sA2[abase + k0 + 1];
      b2.x = sW2[ka * N + bbase];
      b2.y = sW2[(ka + 1) * N + bbase];
      c = __builtin_amdgcn_wmma_f32_16x16x4_f32(false, a2, false, b2,
                                                (short)0, c, false, false);
    }
  }

  const float bval = bias ? bias[bbase] : 0.0f;
#pragma unroll
  for (int r = 0; r < 8; ++r) {
    int grow = mbase + r + 8 * half;
    if (grow < M) {
      float v = c[r] + bval;
      if (RELU) v = fmaxf(v, 0.0f);
      out[(size_t)grow * ldo + col_off + bbase] = v;
    }
  }
}

// ---------------------------------------------------------------------------
// Edge kernels
// ---------------------------------------------------------------------------
__global__ void deg_kernel(const int* __restrict__ dst, float* __restrict__ deg,
                           int nE) {
  int e = blockIdx.x * blockDim.x + threadIdx.x;
  if (e < nE) atomicAdd(&deg[dst[e]], 1.0f);
}

// 16 threads per edge, each moves one float4 of the 64-feature row.
__global__ void scatter_kernel(const float* __restrict__ xin,
                               const int* __restrict__ src,
                               const int* __restrict__ dst,
                               float* __restrict__ agg, int nE) {
  int t = blockIdx.x * blockDim.x + threadIdx.x;
  int e = t >> 4;
  int q = t & 15;
  if (e < nE) {
    int s = src[e], d = dst[e];
    float4 v = ((const float4*)(xin + (size_t)s * HID))[q];
    float* ap = agg + (size_t)d * HID + q * 4;
    atomicAdd(ap + 0, v.x);
    atomicAdd(ap + 1, v.y);
    atomicAdd(ap + 2, v.z);
    atomicAdd(ap + 3, v.w);
  }
}

// ---------------------------------------------------------------------------
// Attention pooling
// ---------------------------------------------------------------------------
__global__ void gate_kernel(const float* __restrict__ xc,
                            const float* __restrict__ gw,
                            const float* __restrict__ gb,
                            float* __restrict__ g, int M) {
  int n = blockIdx.x * blockDim.x + threadIdx.x;
  if (n < M) {
    float s = gb[0];
    const float* row = xc + (size_t)n * GDIM;
#pragma unroll
    for (int i = 0; i < GDIM; ++i) s += row[i] * gw[i];
    g[n] = s;
  }
}

__global__ void pool_init_kernel(float* __restrict__ m, float* __restrict__ d,
                                 float* __restrict__ pooled) {
  int t = threadIdx.x;
  if (t < NGRAPH) { m[t] = -3.0e38f; d[t] = 0.0f; }
  if (t < NGRAPH * GDIM) pooled[t] = 0.0f;
}

__device__ inline void atomicMaxF(float* addr, float value) {
  float old = *addr;
  while (old < value) {
    int assumed = __float_as_int(old);
    int prev = atomicCAS((int*)addr, assumed, __float_as_int(value));
    if (prev == assumed) break;
    old = __int_as_float(prev);
  }
}

__global__ void segmax_kernel(const float* __restrict__ g,
                              const int* __restrict__ batch,
                              float* __restrict__ m, int M) {
  int n = blockIdx.x * blockDim.x + threadIdx.x;
  if (n < M) atomicMaxF(&m[batch[n]], g[n]);
}

__global__ void expsum_kernel(const float* __restrict__ g,
                              const int* __restrict__ batch,
                              const float* __restrict__ m,
                              float* __restrict__ eexp,
                              float* __restrict__ d, int M) {
  int n = blockIdx.x * blockDim.x + threadIdx.x;
  if (n < M) {
    int b = batch[n];
    float e = __expf(g[n] - m[b]);
    eexp[n] = e;
    atomicAdd(&d[b], e);
  }
}

// 8 nodes/block x 32 feats; LDS 16x32 accumulator, then one atomic per slot.
__global__ void pool_kernel(const float* __restrict__ eexp,
                            const float* __restrict__ d,
                            const float* __restrict__ v,
                            const int* __restrict__ batch,
                            float* __restrict__ pooled, int M) {
  __shared__ float acc[NGRAPH * GDIM];
  for (int i = threadIdx.x; i < NGRAPH * GDIM; i += blockDim.x) acc[i] = 0.0f;
  __syncthreads();
  int n = blockIdx.x * 8 + (threadIdx.x >> 5);
  int f = threadIdx.x & 31;
  if (n < M) {
    int b = batch[n];
    float a = eexp[n] / d[b];
    atomicAdd(&acc[b * GDIM + f], a * v[(size_t)n * GDIM + f]);
  }
  __syncthreads();
  for (int i = threadIdx.x; i < NGRAPH * GDIM; i += blockDim.x)
    if (acc[i] != 0.0f) atomicAdd(&pooled[i], acc[i]);
}

// 32 threads: thread t -> (graph t>>1, class t&1). out[0:32]=softmax, [32:64]=logits.
__global__ void head_kernel(const float* __restrict__ pooled,
                            const float* __restrict__ ow,
                            const float* __restrict__ ob,
                            float* __restrict__ out) {
  __shared__ float lg[NGRAPH * 2];
  int t = threadIdx.x;
  int gph = t >> 1, c = t & 1;
  float s = ob[c];
#pragma unroll
  for (int i = 0; i < GDIM; ++i) s += pooled[gph * GDIM + i] * ow[i * 2 + c];
  lg[t] = s;
  __syncthreads();
  float l0 = lg[gph * 2 + 0], l1 = lg[gph * 2 + 1];
  float mx = fmaxf(l0, l1);
  float e0 = __expf(l0 - mx), e1 = __expf(l1 - mx);
  out[t] = ((c == 0) ? e0 : e1) / (e0 + e1);
  out[NGRAPH * 2 + t] = s;
}

// ---------------------------------------------------------------------------
// Launch
// ---------------------------------------------------------------------------
extern "C" void kernel_launch(void* const* d_in, const int* in_sizes, int n_in,
                              void* d_out, int out_size, void* d_ws, size_t ws_size,
                              hipStream_t stream) {
  const float* x      = (const float*)d_in[0];
  const int*   ei     = (const int*)d_in[1];
  const int*   src    = ei;
  const int*   dst    = ei + NEDGES;
  const int*   batch  = (const int*)d_in[2];
  const float* lin0_w = (const float*)d_in[3];
  const float* lin0_b = (const float*)d_in[4];
  const float* s1_lw  = (const float*)d_in[5];
  const float* s1_lb  = (const float*)d_in[6];
  const float* s1_rw  = (const float*)d_in[7];
  const float* s2_lw  = (const float*)d_in[8];
  const float* s2_lb  = (const float*)d_in[9];
  const float* s2_rw  = (const float*)d_in[10];
  const float* emb0_w = (const float*)d_in[11];
  const float* emb0_b = (const float*)d_in[12];
  const float* emb1_w = (const float*)d_in[13];
  const float* emb1_b = (const float*)d_in[14];
  const float* emb2_w = (const float*)d_in[15];
  const float* emb2_b = (const float*)d_in[16];
  const float* merge_w= (const float*)d_in[17];
  const float* merge_b= (const float*)d_in[18];
  const float* gate_w = (const float*)d_in[19];
  const float* gate_b = (const float*)d_in[20];
  const float* v_w    = (const float*)d_in[21];
  const float* v_b    = (const float*)d_in[22];
  const float* out_w  = (const float*)d_in[23];
  const float* out_b  = (const float*)d_in[24];
  float* out = (float*)d_out;

  // Workspace layout (floats), with lifetime-based reuse.
  float* ws   = (float*)d_ws;
  float* x0   = ws;                        // NNODES*64
  float* x1v  = x0  + (size_t)NNODES * HID;
  float* x2v  = x1v + (size_t)NNODES * HID;
  float* agg  = x2v + (size_t)NNODES * HID;   // NNODES*64 (dead after sage2)
  float* econ = agg;                          // NNODES*96 (overlaps agg extent)
  float* deg  = econ + (size_t)NNODES * 96;
  float* g    = deg  + NNODES;
  float* eexp = g    + NNODES;
  float* msum = eexp + NNODES;                // 16
  float* dsum = msum + NGRAPH;                // 16
  float* pooled = dsum + NGRAPH;              // 512
  float* xc = x0;                             // reuse: x0 dead after e0
  float* vv = x1v;                            // reuse: x1 dead after e2

  const int mtiles = (NNODES + 15) / 16;      // 6250
  const dim3 g1((NEDGES + 255) / 256), b1(256);
  const dim3 gs((NEDGES * 16 + 255) / 256), bs(256);
  const dim3 gn((NNODES + 255) / 256), bn(256);

  // degree (shared by both SAGE layers)
  hipMemsetAsync(deg, 0, (size_t)NNODES * 4, stream);
  deg_kernel<<<g1, b1, 0, stream>>>(dst, deg, NEDGES);

  // x0 = x @ lin0_w + lin0_b
  wmma_gemm_kernel<64, 64, false, false, false><<<mtiles, 128, 0, stream>>>(
      x, nullptr, lin0_w, nullptr, nullptr, lin0_b, x0, NNODES, 64, 0);

  // SAGE 1: x1 = mean(agg(x0)) @ s1_lw + x0 @ s1_rw + s1_lb
  hipMemsetAsync(agg, 0, (size_t)NNODES * HID * 4, stream);
  scatter_kernel<<<gs, bs, 0, stream>>>(x0, src, dst, agg, NEDGES);
  wmma_gemm_kernel<64, 64, true, true, false><<<mtiles, 128, 0, stream>>>(
      agg, deg, s1_lw, x0, s1_rw, s1_lb, x1v, NNODES, 64, 0);

  // SAGE 2
  hipMemsetAsync(agg, 0, (size_t)NNODES * HID * 4, stream);
  scatter_kernel<<<gs, bs, 0, stream>>>(x1v, src, dst, agg, NEDGES);
  wmma_gemm_kernel<64, 64, true, true, false><<<mtiles, 128, 0, stream>>>(
      agg, deg, s2_lw, x1v, s2_rw, s2_lb, x2v, NNODES, 64, 0);

  // Embedding heads, written directly into the concat buffer (ldo=96)
  wmma_gemm_kernel<64, 32, false, false, true><<<mtiles, 64, 0, stream>>>(
      x0, nullptr, emb0_w, nullptr, nullptr, emb0_b, econ, NNODES, 96, 0);
  wmma_gemm_kernel<64, 32, false, false, true><<<mtiles, 64, 0, stream>>>(
      x1v, nullptr, emb1_w, nullptr, nullptr, emb1_b, econ, NNODES, 96, 32);
  wmma_gemm_kernel<64, 32, false, false, true><<<mtiles, 64, 0, stream>>>(
      x2v, nullptr, emb2_w, nullptr, nullptr, emb2_b, econ, NNODES, 96, 64);

  // xc = relu(concat @ merge_w + merge_b)    (K=96)
  wmma_gemm_kernel<96, 32, false, false, true><<<mtiles, 64, 0, stream>>>(
      econ, nullptr, merge_w, nullptr, nullptr, merge_b, xc, NNODES, 32, 0);

  // gate scores + values
  gate_kernel<<<gn, bn, 0, stream>>>(xc, gate_w, gate_b, g, NNODES);
  wmma_gemm_kernel<32, 32, false, false, false><<<mtiles, 64, 0, stream>>>(
      xc, nullptr, v_w, nullptr, nullptr, v_b, vv, NNODES, 32, 0);

  // per-graph softmax attention pooling
  pool_init_kernel<<<1, 512, 0, stream>>>(msum, dsum, pooled);
  segmax_kernel<<<gn, bn, 0, stream>>>(g, batch, msum, NNODES);
  expsum_kernel<<<gn, bn, 0, stream>>>(g, batch, msum, eexp, dsum, NNODES);
  pool_kernel<<<(NNODES + 7) / 8, 256, 0, stream>>>(eexp, dsum, vv, batch,
                                                    pooled, NNODES);

  // logits + softmax -> out[0:32] softmax, out[32:64] logits
  head_kernel<<<1, 32, 0, stream>>>(pooled, out_w, out_b, out);
}